// CBAMBlock_8272107012661
// MI455X (gfx1250) — compile-verified
//
#include <hip/hip_runtime.h>
#include <math.h>

// ---------------------------------------------------------------------------
// CBAM fused pipeline for MI455X (gfx1250).
//  x:[32,56,56,256] f32.  B=32, H=W=56, C=256, R=16, K=7.
//  Memory-bound: x = 103MB fits in 192MB L2 -> re-read x from L2 instead of
//  materializing xr (saves 206MB of HBM traffic).  MLP done with
//  v_wmma_f32_16x16x4_f32 (fp32, wave32).  Final output stored non-temporal
//  so the 103MB of writes do not evict x from L2 mid-kernel.
// ---------------------------------------------------------------------------

typedef __attribute__((ext_vector_type(2))) float v2f;
typedef __attribute__((ext_vector_type(4))) float v4f;
typedef __attribute__((ext_vector_type(8))) float v8f;

#define BATCH   32
#define HW      3136      // 56*56
#define CH      256
#define HID     16
#define NSPLIT  14
#define PIX_PER_SPLIT 224 // 3136/14
#define NPIX    100352    // 32*3136

__device__ __forceinline__ float sigmoidf_(float v) {
    return 1.0f / (1.0f + expf(-v));
}

// ---------------------------------------------------------------------------
// Kernel 1: partial avg/max pooling over H,W.  448 blocks x 256 thr.
// Thread c owns channel c -> fully coalesced 1KB line per pixel.
// First touch of x: fills L2 (RT policy) for the later passes.
// ---------------------------------------------------------------------------
__global__ void k_pool_partial(const float* __restrict__ x,
                               float* __restrict__ partial) {
    const int b = blockIdx.x / NSPLIT;
    const int s = blockIdx.x % NSPLIT;
    const int c = threadIdx.x;
    const float* xp = x + ((size_t)b * HW + (size_t)s * PIX_PER_SPLIT) * CH + c;
    float sum = 0.0f, mx = -3.402823466e38f;
    #pragma unroll 4
    for (int p = 0; p < PIX_PER_SPLIT; ++p) {
        float v = xp[(size_t)p * CH];
        sum += v;
        mx = fmaxf(mx, v);
    }
    float* dst = partial + ((size_t)(b * NSPLIT + s)) * (2 * CH);
    dst[c]       = sum;
    dst[CH + c]  = mx;
}

// ---------------------------------------------------------------------------
// Kernel 2: fold the 14 partials -> pooled P[64][256]
//   rows 0..31  = avg pool (mean), rows 32..63 = max pool
// ---------------------------------------------------------------------------
__global__ void k_pool_reduce(const float* __restrict__ partial,
                              float* __restrict__ P) {
    const int b = blockIdx.x;
    const int c = threadIdx.x;
    float sum = 0.0f, mx = -3.402823466e38f;
    for (int s = 0; s < NSPLIT; ++s) {
        const float* src = partial + ((size_t)(b * NSPLIT + s)) * (2 * CH);
        sum += src[c];
        mx = fmaxf(mx, src[CH + c]);
    }
    P[b * CH + c]            = sum * (1.0f / (float)HW);
    P[(BATCH + b) * CH + c]  = mx;
}

// ---------------------------------------------------------------------------
// Kernel 3: shared MLP via WMMA f32 16x16x4 + sigmoid -> ca[32][256].
// Single workgroup, 8 waves.  Layer1: P[64,256]@w1[256,16] (+b1, relu).
// Layer2: tiles paired (avg row-tile, max row-tile) so each wave emits
// sigmoid(Da + Db + 2*b2) directly.
// Fragment layouts per ISA 7.12.2:
//   A 16x4 f32 : m = lane%16 ; K pair = lanes<16 ? {0,1} : {2,3}
//   B 4x16 f32 : n = lane%16 ; K pair = lanes<16 ? {0,1} : {2,3}
//   C/D        : vgpr r -> (M = r + (lane<16?0:8), N = lane%16)
// ---------------------------------------------------------------------------
__global__ void k_mlp_wmma(const float* __restrict__ P,
                           const float* __restrict__ w1,
                           const float* __restrict__ b1,
                           const float* __restrict__ w2,
                           const float* __restrict__ b2,
                           float* __restrict__ ca) {
    __shared__ float H1[64][17];   // pitch 17 to avoid LDS bank conflicts

    const int wave = threadIdx.x >> 5;
    const int lane = threadIdx.x & 31;
    const int m16  = lane & 15;
    const int hi   = lane >> 4;       // 0: lanes 0-15, 1: lanes 16-31
    const int k0   = hi * 2;

    // ---- Layer 1: 4 M-tiles (64 rows), N=16, K=256 (64 wmma steps) ----
    if (wave < 4) {
        const int mt = wave;
        v8f acc = {};
        const float* arow = P + (size_t)(mt * 16 + m16) * CH;
        #pragma unroll 8
        for (int kk = 0; kk < CH / 4; ++kk) {
            const int kb = kk * 4;
            v2f a, bfr;
            a.x   = arow[kb + k0];
            a.y   = arow[kb + k0 + 1];
            bfr.x = w1[(kb + k0) * HID + m16];
            bfr.y = w1[(kb + k0 + 1) * HID + m16];
            acc = __builtin_amdgcn_wmma_f32_16x16x4_f32(
                false, a, false, bfr, (short)0, acc, false, false);
        }
        const float bias = b1[m16];
        #pragma unroll
        for (int r = 0; r < 8; ++r) {
            const int m = mt * 16 + r + hi * 8;
            const float v = acc[r] + bias;
            H1[m][m16] = v > 0.0f ? v : 0.0f;
        }
    }
    __syncthreads();

    // ---- Layer 2: pairs (mt, mt+2) over 16 N-tiles; 4 pairs per wave ----
    #pragma unroll
    for (int p = 0; p < 4; ++p) {
        const int pid = wave * 4 + p;     // 0..31
        const int mt  = pid >> 4;         // 0..1  (avg rows; +32 is max rows)
        const int nt  = pid & 15;         // 0..15
        v8f da = {}, db = {};
        #pragma unroll
        for (int kk = 0; kk < HID / 4; ++kk) {
            const int kb = kk * 4;
            v2f bfr, aa, ab;
            bfr.x = w2[(kb + k0) * CH + nt * 16 + m16];
            bfr.y = w2[(kb + k0 + 1) * CH + nt * 16 + m16];
            aa.x = H1[mt * 16 + m16][kb + k0];
            aa.y = H1[mt * 16 + m16][kb + k0 + 1];
            ab.x = H1[32 + mt * 16 + m16][kb + k0];
            ab.y = H1[32 + mt * 16 + m16][kb + k0 + 1];
            da = __builtin_amdgcn_wmma_f32_16x16x4_f32(
                false, aa, false, bfr, (short)0, da, false, false);
            db = __builtin_amdgcn_wmma_f32_16x16x4_f32(
                false, ab, false, bfr, (short)0, db, false, false);
        }
        const int n = nt * 16 + m16;
        const float b2n = b2[n];
        #pragma unroll
        for (int r = 0; r < 8; ++r) {
            const int m = mt * 16 + r + hi * 8;     // batch index 0..31
            ca[m * CH + n] = sigmoidf_(da[r] + db[r] + 2.0f * b2n);
        }
    }
}

// ---------------------------------------------------------------------------
// Kernel 4: per-pixel channel mean/max of x*ca (xr never materialized).
// One wave32 per pixel; float4 loads (global_load_b128, L2-hot x);
// shfl_xor wave32 reduction.
// ---------------------------------------------------------------------------
__global__ void k_spatial_pool(const float* __restrict__ x,
                               const float* __restrict__ ca,
                               float* __restrict__ spm,
                               float* __restrict__ spx) {
    const int wave = threadIdx.x >> 5;
    const int lane = threadIdx.x & 31;
    const int pid  = blockIdx.x * 8 + wave;        // < 100352
    const int b    = pid / HW;

    const v4f* xp  = (const v4f*)(x + (size_t)pid * CH);
    const v4f* cap = (const v4f*)(ca + (size_t)b * CH);

    float s = 0.0f, m = -3.402823466e38f;
    #pragma unroll
    for (int j = 0; j < 2; ++j) {
        const int i = lane + j * 32;               // 64 float4 = 256 ch
        v4f v = xp[i] * cap[i];
        s += v.x + v.y + v.z + v.w;
        m = fmaxf(m, fmaxf(fmaxf(v.x, v.y), fmaxf(v.z, v.w)));
    }
    #pragma unroll
    for (int off = 16; off > 0; off >>= 1) {
        s += __shfl_xor(s, off, 32);
        m = fmaxf(m, __shfl_xor(m, off, 32));
    }
    if (lane == 0) {
        spm[pid] = s * (1.0f / (float)CH);
        spx[pid] = m;
    }
}

// ---------------------------------------------------------------------------
// Kernel 5: 7x7 SAME conv over [mean,max] + sigmoid -> sa[B*H*W].  Tiny.
// ---------------------------------------------------------------------------
__global__ void k_conv_sa(const float* __restrict__ spm,
                          const float* __restrict__ spx,
                          const float* __restrict__ cw,   // [7][7][2]
                          float* __restrict__ sa) {
    const int pid = blockIdx.x * blockDim.x + threadIdx.x; // < 100352
    const int b   = pid / HW;
    const int rem = pid % HW;
    const int h   = rem / 56;
    const int w   = rem % 56;
    float acc = 0.0f;
    #pragma unroll
    for (int kh = 0; kh < 7; ++kh) {
        const int ih = h + kh - 3;
        if (ih < 0 || ih >= 56) continue;
        #pragma unroll
        for (int kw = 0; kw < 7; ++kw) {
            const int iw = w + kw - 3;
            if (iw < 0 || iw >= 56) continue;
            const int q = b * HW + ih * 56 + iw;
            const int wi = (kh * 7 + kw) * 2;
            acc += spm[q] * cw[wi] + spx[q] * cw[wi + 1];
        }
    }
    sa[pid] = sigmoidf_(acc);
}

// ---------------------------------------------------------------------------
// Kernel 6: out = x * ca[b,c] * sa[b,h,w]   (float4 b128 traffic).
// x is read from L2 (RT); out is write-once -> NON-TEMPORAL store so the
// 103MB of writebacks rinse through L2 without evicting x while we still
// need it.
// ---------------------------------------------------------------------------
__global__ void k_apply(const float* __restrict__ x,
                        const float* __restrict__ ca,
                        const float* __restrict__ sa,
                        float* __restrict__ out) {
    const int idx = blockIdx.x * blockDim.x + threadIdx.x;  // float4 index
    const int pix = idx >> 6;        // /64 float4s per pixel
    const int c4  = idx & 63;
    const int b   = pix / HW;
    v4f xv = ((const v4f*)x)[idx];
    v4f cv = ((const v4f*)ca)[b * 64 + c4];
    float s = sa[pix];
    v4f o = xv * cv * s;
    __builtin_nontemporal_store(o, (v4f*)out + idx);
}

// ---------------------------------------------------------------------------
extern "C" void kernel_launch(void* const* d_in, const int* in_sizes, int n_in,
                              void* d_out, int out_size, void* d_ws, size_t ws_size,
                              hipStream_t stream) {
    const float* x      = (const float*)d_in[0];
    const float* w1     = (const float*)d_in[1];
    const float* b1     = (const float*)d_in[2];
    const float* w2     = (const float*)d_in[3];
    const float* b2     = (const float*)d_in[4];
    const float* conv_w = (const float*)d_in[5];
    float* out = (float*)d_out;

    // workspace layout (floats)
    float* ws      = (float*)d_ws;
    float* partial = ws;                                  // 32*14*2*256 = 229376
    float* P       = partial + BATCH * NSPLIT * 2 * CH;   // 64*256      = 16384
    float* ca      = P + 64 * CH;                         // 32*256      = 8192
    float* spm     = ca + BATCH * CH;                     // 100352
    float* spx     = spm + NPIX;                          // 100352
    float* sa      = spx + NPIX;                          // 100352
    (void)in_sizes; (void)n_in; (void)out_size; (void)ws_size;

    k_pool_partial<<<BATCH * NSPLIT, CH, 0, stream>>>(x, partial);
    k_pool_reduce<<<BATCH, CH, 0, stream>>>(partial, P);
    k_mlp_wmma<<<1, 256, 0, stream>>>(P, w1, b1, w2, b2, ca);
    k_spatial_pool<<<NPIX / 8, 256, 0, stream>>>(x, ca, spm, spx);
    k_conv_sa<<<NPIX / 256, 256, 0, stream>>>(spm, spx, conv_w, sa);
    k_apply<<<(NPIX * CH / 4) / 256, 256, 0, stream>>>(x, ca, sa, out);
}